// AttentionHead_88862873355093
// MI455X (gfx1250) — compile-verified
//
#include <hip/hip_runtime.h>

typedef float v2f __attribute__((ext_vector_type(2)));
typedef float v8f __attribute__((ext_vector_type(8)));

#define BDIM 4
#define TDIM 4096
#define MD 64
#define HD 16
#define SEG 1024                    // keys per partial segment
#define NSEG (TDIM / SEG)           // 4
#define NTILES (BDIM * TDIM / 16)   // 1024 query tiles of 16 rows
#define PART_STRIDE 288             // 256 floats O + 16 m + 16 l

#define NEG_INF (-__builtin_inff())

// One fp32 WMMA step: D(16x16) = A(16x4) * B(4x16) + C, chained for larger K.
static __device__ __forceinline__ v8f wmma4(v2f a, v2f b, v8f c) {
    return __builtin_amdgcn_wmma_f32_16x16x4_f32(
        /*neg_a=*/false, a, /*neg_b=*/false, b,
        /*c_mod=*/(short)0, c, /*reuse_a=*/false, /*reuse_b=*/false);
}

// ---------------------------------------------------------------------------
// Kernel 1: QKV projection. grid = B*T/16, 32 threads (1 wave).
// 16-row tile of q,k,v = idx @ W^T + b via 16 chained f32 WMMAs (K=64).
// ---------------------------------------------------------------------------
__global__ __launch_bounds__(32) void qkv_proj(
    const float* __restrict__ idx,
    const float* __restrict__ Wq, const float* __restrict__ bq,
    const float* __restrict__ Wk, const float* __restrict__ bk,
    const float* __restrict__ Wv, const float* __restrict__ bv,
    float* __restrict__ qo, float* __restrict__ ko, float* __restrict__ vo)
{
    const int lane = threadIdx.x;
    const int l16  = lane & 15;
    const int half = lane >> 4;
    const int row0 = blockIdx.x * 16;

    v2f a[16];
    const float* ap = idx + (size_t)(row0 + l16) * MD + half * 2;
#pragma unroll
    for (int kc = 0; kc < 16; ++kc) {
        a[kc][0] = ap[kc * 4 + 0];
        a[kc][1] = ap[kc * 4 + 1];
    }

    auto proj = [&](const float* __restrict__ W, const float* __restrict__ bias,
                    float* __restrict__ outp) {
        v8f acc = {0.f, 0.f, 0.f, 0.f, 0.f, 0.f, 0.f, 0.f};
        const float* wp = W + (size_t)l16 * MD + half * 2;
#pragma unroll
        for (int kc = 0; kc < 16; ++kc) {
            v2f bv2; bv2[0] = wp[kc * 4 + 0]; bv2[1] = wp[kc * 4 + 1];
            acc = wmma4(a[kc], bv2, acc);
        }
        const float bb = bias[l16];
#pragma unroll
        for (int r = 0; r < 8; ++r)
            outp[(size_t)(row0 + r + 8 * half) * HD + l16] = acc[r] + bb;
    };

    proj(Wq, bq, qo);
    proj(Wk, bk, ko);
    proj(Wv, bv, vo);
}

// ---------------------------------------------------------------------------
// Kernel 2: partial flash attention over one key segment.
// grid = NTILES*NSEG, 32 threads. Block = (query tile tt, segment s).
// Computes S^T = K*Q^T so each lane owns one query's scores: softmax needs
// only register-local reductions + one xor-16 shuffle; exp(S^T) is converted
// to the P A-layout with a single xor-16 exchange. No LDS, no barriers.
// Writes partial (O[16x16], m[16], l[16]) to workspace.
// ---------------------------------------------------------------------------
__global__ __launch_bounds__(32) void attn_partial(
    const float* __restrict__ q, const float* __restrict__ k,
    const float* __restrict__ v, float* __restrict__ part)
{
    const int lane = threadIdx.x;
    const int l16  = lane & 15;
    const int half = lane >> 4;

    const int tt = blockIdx.x / NSEG;          // global query tile id
    const int s  = blockIdx.x % NSEG;          // key segment id

    const int tpb = TDIM / 16;
    const int b   = tt / tpb;
    const int rb  = (tt % tpb) * 16;           // batch-local first query row

    float* pb = part + (size_t)(tt * NSEG + s) * PART_STRIDE;

    const int kstart = s * SEG;
    const int kend   = (kstart + SEG < rb + 16) ? (kstart + SEG) : (rb + 16);

    if (kstart >= kend) {                      // uniform: no keys -> neutral
#pragma unroll
        for (int r = 0; r < 8; ++r) pb[r * 32 + lane] = 0.0f;
        if (lane < 16) { pb[256 + lane] = NEG_INF; pb[272 + lane] = 0.0f; }
        return;
    }

    const float* qbp = q + (size_t)b * TDIM * HD;
    const float* kbp = k + (size_t)b * TDIM * HD;
    const float* vbp = v + (size_t)b * TDIM * HD;

    // Q tile as WMMA B operand (per-lane pattern identical to A-layout).
    v2f qa[4];
    {
        const float* p = qbp + (size_t)(rb + l16) * HD + half * 2;
#pragma unroll
        for (int kc = 0; kc < 4; ++kc) { qa[kc][0] = p[kc*4]; qa[kc][1] = p[kc*4+1]; }
    }

    v8f o = {0.f, 0.f, 0.f, 0.f, 0.f, 0.f, 0.f, 0.f};
    float m = NEG_INF;
    float l = 0.0f;
    const float scale = 0.25f;                 // 1/sqrt(HD)
    const int   qrow  = rb + l16;              // this lane's query row

    for (int kt0 = kstart; kt0 < kend; kt0 += 16) {
        // K tile as A operand; V tile in B-layout.
        v2f kt[4], vt[4];
        {
            const float* p = kbp + (size_t)(kt0 + l16) * HD + half * 2;
#pragma unroll
            for (int kc = 0; kc < 4; ++kc) { kt[kc][0] = p[kc*4]; kt[kc][1] = p[kc*4+1]; }
        }
        {
            const float* p = vbp + (size_t)(kt0 + half * 2) * HD + l16;
#pragma unroll
            for (int kc = 0; kc < 4; ++kc) {
                vt[kc][0] = p[(kc * 4 + 0) * HD];
                vt[kc][1] = p[(kc * 4 + 1) * HD];
            }
        }

        // S^T = K * Q^T : vgpr r = key (kt0 + r + 8*half), lane = query l16.
        v8f st = {0.f, 0.f, 0.f, 0.f, 0.f, 0.f, 0.f, 0.f};
#pragma unroll
        for (int kc = 0; kc < 4; ++kc) st = wmma4(kt[kc], qa[kc], st);

        // Mask + scale (faithful: upper triangle OR exact zero -> -inf),
        // then online softmax with per-lane scalar stats.
        float pe[8];
        float bm = NEG_INF;
#pragma unroll
        for (int r = 0; r < 8; ++r) {
            const int key = kt0 + r + 8 * half;
            float sv = st[r] * scale;
            sv = (key <= qrow && sv != 0.0f) ? sv : NEG_INF;
            pe[r] = sv;
            bm = fmaxf(bm, sv);
        }
        bm = fmaxf(bm, __shfl_xor(bm, 16, 32));   // full 16-key row max

        const float mn    = fmaxf(m, bm);
        const float alpha = (mn == NEG_INF) ? 1.0f : __expf(m - mn);
        float bl = 0.0f;
#pragma unroll
        for (int r = 0; r < 8; ++r) {
            pe[r] = (mn == NEG_INF) ? 0.0f : __expf(pe[r] - mn);
            bl += pe[r];
        }
        bl += __shfl_xor(bl, 16, 32);             // full 16-key row sum
        l = l * alpha + bl;
        m = mn;

        // Rescale O (D-layout rows = queries): broadcast alpha per row.
#pragma unroll
        for (int r = 0; r < 8; ++r) o[r] *= __shfl(alpha, r + 8 * half, 32);

        // Rearrange pe (S^T D-layout) -> P A-layout: one xor-16 exchange.
        const float snd0 = half ? pe[0] : pe[2];
        const float snd1 = half ? pe[1] : pe[3];
        const float snd2 = half ? pe[4] : pe[6];
        const float snd3 = half ? pe[5] : pe[7];
        const float rcv0 = __shfl_xor(snd0, 16, 32);
        const float rcv1 = __shfl_xor(snd1, 16, 32);
        const float rcv2 = __shfl_xor(snd2, 16, 32);
        const float rcv3 = __shfl_xor(snd3, 16, 32);

        v2f pa[4];
        pa[0][0] = half ? rcv0  : pe[0];  pa[0][1] = half ? rcv1  : pe[1];
        pa[1][0] = half ? rcv2  : pe[4];  pa[1][1] = half ? rcv3  : pe[5];
        pa[2][0] = half ? pe[2] : rcv0;   pa[2][1] = half ? pe[3] : rcv1;
        pa[3][0] = half ? pe[6] : rcv2;   pa[3][1] = half ? pe[7] : rcv3;

        // O += P * V (4 chained f32 WMMAs).
#pragma unroll
        for (int kc = 0; kc < 4; ++kc) o = wmma4(pa[kc], vt[kc], o);
    }

    // Write partial: O in D-layout, stats from lanes 0..15.
#pragma unroll
    for (int r = 0; r < 8; ++r) pb[r * 32 + lane] = o[r];
    if (lane < 16) { pb[256 + lane] = m; pb[272 + lane] = l; }
}

// ---------------------------------------------------------------------------
// Kernel 3: merge segment partials, divide by l, project with Wp + bp.
// grid = NTILES, 32 threads.
// ---------------------------------------------------------------------------
__global__ __launch_bounds__(32) void attn_combine(
    const float* __restrict__ part, const float* __restrict__ Wp,
    const float* __restrict__ bp, float* __restrict__ out)
{
    __shared__ float sP[16][20];

    const int lane = threadIdx.x;
    const int l16  = lane & 15;
    const int half = lane >> 4;
    const int tt   = blockIdx.x;

    const float* pb0 = part + (size_t)tt * NSEG * PART_STRIDE;

    v8f av;
#pragma unroll
    for (int r = 0; r < 8; ++r) {
        const int row = r + 8 * half;
        float ms[NSEG];
        float mx = NEG_INF;
#pragma unroll
        for (int s = 0; s < NSEG; ++s) {
            ms[s] = pb0[s * PART_STRIDE + 256 + row];
            mx = fmaxf(mx, ms[s]);
        }
        float osum = 0.0f, lsum = 0.0f;
#pragma unroll
        for (int s = 0; s < NSEG; ++s) {
            const float f = (mx == NEG_INF) ? 1.0f : __expf(ms[s] - mx);
            lsum += f * pb0[s * PART_STRIDE + 272 + row];
            osum += f * pb0[s * PART_STRIDE + r * 32 + lane];
        }
        av[r] = osum / lsum;   // all-masked row: 0/0 -> NaN (faithful)
    }

    // Transpose av: D-layout -> A-layout through LDS (once per tile).
#pragma unroll
    for (int r = 0; r < 8; ++r) sP[r + 8 * half][l16] = av[r];
    __syncthreads();

    v2f aa[4];
#pragma unroll
    for (int kc = 0; kc < 4; ++kc) {
        aa[kc][0] = sP[l16][kc * 4 + half * 2 + 0];
        aa[kc][1] = sP[l16][kc * 4 + half * 2 + 1];
    }

    // out = av @ Wp^T + bp : 4 column blocks x 4 chained WMMAs each.
#pragma unroll
    for (int j = 0; j < 4; ++j) {
        const float bb = bp[j * 16 + l16];
        v8f d = {bb, bb, bb, bb, bb, bb, bb, bb};
        const float* wp = Wp + (size_t)(j * 16 + l16) * HD + half * 2;
#pragma unroll
        for (int kc = 0; kc < 4; ++kc) {
            v2f wv2; wv2[0] = wp[kc * 4 + 0]; wv2[1] = wp[kc * 4 + 1];
            d = wmma4(aa[kc], wv2, d);
        }
#pragma unroll
        for (int r = 0; r < 8; ++r)
            out[((size_t)tt * 16 + r + 8 * half) * MD + j * 16 + l16] = d[r];
    }
}

// ---------------------------------------------------------------------------
extern "C" void kernel_launch(void* const* d_in, const int* in_sizes, int n_in,
                              void* d_out, int out_size, void* d_ws, size_t ws_size,
                              hipStream_t stream) {
    // setup_inputs() order: idx, Wk, bk, Wq, bq, Wv, bv, Wp, bp
    const float* idx = (const float*)d_in[0];
    const float* Wk  = (const float*)d_in[1];
    const float* bk  = (const float*)d_in[2];
    const float* Wq  = (const float*)d_in[3];
    const float* bq  = (const float*)d_in[4];
    const float* Wv  = (const float*)d_in[5];
    const float* bv  = (const float*)d_in[6];
    const float* Wp  = (const float*)d_in[7];
    const float* bp  = (const float*)d_in[8];
    float* out = (float*)d_out;

    const size_t BT = (size_t)BDIM * TDIM;           // 16384 rows
    float* qws  = (float*)d_ws;                      // [BT, HD]   1 MB
    float* kws  = qws + BT * HD;                     // [BT, HD]   1 MB
    float* vws  = kws + BT * HD;                     // [BT, HD]   1 MB
    float* pws  = vws + BT * HD;                     // partials  ~4.7 MB

    qkv_proj<<<NTILES, 32, 0, stream>>>(idx, Wq, bq, Wk, bk, Wv, bv,
                                        qws, kws, vws);
    attn_partial<<<NTILES * NSEG, 32, 0, stream>>>(qws, kws, vws, pws);
    attn_combine<<<NTILES, 32, 0, stream>>>(pws, Wp, bp, out);
}